// GraphActorCritic_28295244546816
// MI455X (gfx1250) — compile-verified
//
#include <hip/hip_runtime.h>
#include <hip/hip_bf16.h>

typedef float v2f __attribute__((ext_vector_type(2)));
typedef float v8f __attribute__((ext_vector_type(8)));

// ---------------- helpers ----------------

__device__ __forceinline__ void atomicMaxFloat(float* addr, float value) {
  unsigned int bits = __float_as_uint(value);
  if ((bits >> 31) == 0) {
    // non-negative float: signed-int max works
    atomicMax((int*)addr, (int)bits);
  } else {
    // negative float: unsigned min picks the larger float
    atomicMin((unsigned int*)addr, bits);
  }
}

__device__ __forceinline__ void load_emb(int id, int pos, const float* __restrict__ table,
                                         float emb[7]) {
  const float* t = table + (size_t)id * 4;
  emb[0] = t[0]; emb[1] = t[1]; emb[2] = t[2]; emb[3] = t[3];
  emb[4] = (pos == 0) ? 1.f : 0.f;
  emb[5] = (pos == 1) ? 1.f : 0.f;
  emb[6] = (pos == 2) ? 1.f : 0.f;
}

// ---------------- init ----------------

__global__ __launch_bounds__(256) void init_k(float* smax, float* denom, float* agg,
                                              float* deg, float* red, float* gsum, int n) {
  int i = blockIdx.x * blockDim.x + threadIdx.x;
  if (i < n) {
    smax[i]  = -__builtin_inff();
    denom[i] = 0.f;
    deg[i]   = 1.f;                   // self loop
    for (int k = 0; k < 7; ++k) agg[(size_t)i * 7 + k] = 0.f;
  }
  if (blockIdx.x == 0) {
    if (threadIdx.x < 8) red[threadIdx.x] = 0.f;
    gsum[threadIdx.x] = 0.f;          // blockDim == 256
  }
}

// ---------------- edge attention ----------------

__global__ __launch_bounds__(256) void edge_scores(
    const int* __restrict__ eid, const int* __restrict__ epos,
    const float* __restrict__ table,
    const float* __restrict__ w1, const float* __restrict__ b1,
    const float* __restrict__ w2, const float* __restrict__ b2,
    const int* __restrict__ src, const int* __restrict__ dst,
    float* __restrict__ score, float* __restrict__ smax, float* __restrict__ deg, int E) {
  int e = blockIdx.x * blockDim.x + threadIdx.x;
  if (e >= E) return;
  float emb[7];
  load_emb(eid[e], epos[e], table, emb);
  float sc = b2[0];
  #pragma unroll
  for (int i = 0; i < 16; ++i) {
    float h = b1[i];
    #pragma unroll
    for (int k = 0; k < 7; ++k) h += emb[k] * w1[k * 16 + i];
    sc += fmaxf(h, 0.f) * w2[i];
  }
  score[e] = sc;
  int s = src[e], d = dst[e];
  atomicMaxFloat(&smax[s], sc);
  atomicMaxFloat(&smax[d], sc);
  atomicAdd(&deg[d], 1.f);            // target-degree for GCN norm
}

__global__ __launch_bounds__(256) void edge_expsum(
    const float* __restrict__ score, const int* __restrict__ src, const int* __restrict__ dst,
    const float* __restrict__ smax, float* __restrict__ denom, int E) {
  int e = blockIdx.x * blockDim.x + threadIdx.x;
  if (e >= E) return;
  float sc = score[e];
  int s = src[e], d = dst[e];
  atomicAdd(&denom[s], expf(sc - smax[s]));
  atomicAdd(&denom[d], expf(sc - smax[d]));
}

__global__ __launch_bounds__(256) void edge_agg(
    const int* __restrict__ eid, const int* __restrict__ epos,
    const float* __restrict__ table,
    const float* __restrict__ score, const int* __restrict__ src, const int* __restrict__ dst,
    const float* __restrict__ smax, const float* __restrict__ denom,
    float* __restrict__ agg, int E) {
  int e = blockIdx.x * blockDim.x + threadIdx.x;
  if (e >= E) return;
  float emb[7];
  load_emb(eid[e], epos[e], table, emb);
  float sc = score[e];
  int s = src[e], d = dst[e];
  float ws = expf(sc - smax[s]) / denom[s];
  float wd = expf(sc - smax[d]) / denom[d];
  #pragma unroll
  for (int k = 0; k < 7; ++k) {
    atomicAdd(&agg[(size_t)s * 7 + k], emb[k] * ws);
    atomicAdd(&agg[(size_t)d * 7 + k], emb[k] * wd);
  }
}

// ---------------- node features ----------------

__global__ __launch_bounds__(256) void reduce_stats(const float* __restrict__ a,
                                                    const float* __restrict__ b,
                                                    float* red, int n) {
  __shared__ float s0[256], s1[256], s2[256], s3[256];
  int t = threadIdx.x;
  int i = blockIdx.x * 256 + t;
  float va = (i < n) ? a[i] : 0.f;
  float vb = (i < n) ? b[i] : 0.f;
  s0[t] = va; s1[t] = va * va; s2[t] = vb; s3[t] = vb * vb;
  __syncthreads();
  for (int st = 128; st > 0; st >>= 1) {
    if (t < st) { s0[t] += s0[t+st]; s1[t] += s1[t+st]; s2[t] += s2[t+st]; s3[t] += s3[t+st]; }
    __syncthreads();
  }
  if (t == 0) {
    atomicAdd(&red[0], s0[0]); atomicAdd(&red[1], s1[0]);
    atomicAdd(&red[2], s2[0]); atomicAdd(&red[3], s3[0]);
  }
}

__global__ __launch_bounds__(256) void node_feats_k(
    const float* __restrict__ status, const float* __restrict__ est,
    const float* __restrict__ degf, const float* __restrict__ agg,
    const float* __restrict__ red, float* __restrict__ nf, int n) {
  int i = blockIdx.x * blockDim.x + threadIdx.x;
  if (i >= n) return;
  float fn = (float)n;
  float me = red[0] / fn, ve = (red[1] - fn * me * me) / (fn - 1.f);
  float se = sqrtf(fmaxf(ve, 0.f));
  float md = red[2] / fn, vd = (red[3] - fn * md * md) / (fn - 1.f);
  float sd = sqrtf(fmaxf(vd, 0.f));
  float e = est[i] - me;  if (se > 1e-8f) e /= se;
  float d = degf[i] - md; if (sd > 1e-8f) d /= sd;
  float* o = nf + (size_t)i * 10;
  o[0] = status[i]; o[1] = e; o[2] = d;
  for (int k = 0; k < 7; ++k) o[3 + k] = agg[(size_t)i * 7 + k];
}

__global__ __launch_bounds__(256) void dinv_k(float* deg, int n) {
  int i = blockIdx.x * blockDim.x + threadIdx.x;
  if (i < n) deg[i] = rsqrtf(deg[i]);   // deg >= 1 always (self loop)
}

// ---------------- GCN layer pieces ----------------

// h1 = nf (N,10) @ W (10,256)
__global__ __launch_bounds__(256) void conv1_transform(const float* __restrict__ nf,
                                                       const float* __restrict__ W,
                                                       float* __restrict__ out, int n) {
  __shared__ float sW[2560];
  for (int i = threadIdx.x; i < 2560; i += 256) sW[i] = W[i];
  __syncthreads();
  int base = blockIdx.x * 8;
  int j = threadIdx.x;
  for (int nn = 0; nn < 8; ++nn) {
    int node = base + nn;
    if (node >= n) break;
    const float* f = nf + (size_t)node * 10;
    float acc = 0.f;
    #pragma unroll
    for (int k = 0; k < 10; ++k) acc += f[k] * sW[k * 256 + j];
    out[(size_t)node * 256 + j] = acc;
  }
}

// out[n][j] = h[n][j]*dinv[n]^2 + bias[j]   (self-loop term + bias)
__global__ __launch_bounds__(256) void self_init(const float* __restrict__ h,
                                                 const float* __restrict__ dinv,
                                                 const float* __restrict__ bias,
                                                 float* __restrict__ out, int n) {
  size_t idx = (size_t)blockIdx.x * blockDim.x + threadIdx.x;
  if (idx >= (size_t)n * 256) return;
  int node = (int)(idx >> 8);
  int j = (int)(idx & 255);
  float di = dinv[node];
  out[idx] = h[idx] * di * di + bias[j];
}

// one wave per edge: out[dst] += h[src] * dinv[src]*dinv[dst]
__global__ __launch_bounds__(256) void gcn_aggregate(
    const float* __restrict__ h, const int* __restrict__ src, const int* __restrict__ dst,
    const float* __restrict__ dinv, float* __restrict__ out, int E) {
  int e = blockIdx.x * 8 + (threadIdx.x >> 5);
  if (e >= E) return;
  int lane = threadIdx.x & 31;
  int s = src[e], d = dst[e];
  float nrm = dinv[s] * dinv[d];
  const float4* hs = (const float4*)(h + (size_t)s * 256);
  float* od = out + (size_t)d * 256;
  #pragma unroll
  for (int i = 0; i < 2; ++i) {
    float4 v = hs[lane * 2 + i];
    int b = (lane * 2 + i) * 4;
    atomicAdd(od + b + 0, v.x * nrm);
    atomicAdd(od + b + 1, v.y * nrm);
    atomicAdd(od + b + 2, v.z * nrm);
    atomicAdd(od + b + 3, v.w * nrm);
  }
}

__global__ __launch_bounds__(256) void relu_k(float* __restrict__ x, size_t n) {
  size_t i = (size_t)blockIdx.x * blockDim.x + threadIdx.x;
  if (i < n) x[i] = fmaxf(x[i], 0.f);
}

// ---------------- WMMA GEMM: C(N,256) = A(N,256) @ W(256,256) [+bias][relu] ----------------
// fp32 WMMA 16x16x4, one wave per 16x16 output tile.

__global__ __launch_bounds__(256) void gemm256_wmma(
    const float* __restrict__ A, const float* __restrict__ W, float* __restrict__ C,
    const float* __restrict__ bias, int n, int do_relu) {
  int wave = threadIdx.x >> 5;            // 0..7
  int lane = threadIdx.x & 31;
  int row0 = blockIdx.x * 16;
  int col0 = (blockIdx.y * 8 + wave) * 16;
  if (row0 >= n) return;
  int m  = lane & 15;
  int hi = lane >> 4;                      // 0: K pair {k,k+1}, 1: {k+2,k+3}
  v8f c = {0.f, 0.f, 0.f, 0.f, 0.f, 0.f, 0.f, 0.f};
  const float* Arow = A + (size_t)(row0 + m) * 256;
  for (int k = 0; k < 256; k += 4) {
    int ka = k + hi * 2;
    v2f a = *(const v2f*)(Arow + ka);                      // A[m][ka], A[m][ka+1]
    v2f b;
    b.x = W[(size_t)ka * 256 + col0 + m];                  // B[ka][n]
    b.y = W[(size_t)(ka + 1) * 256 + col0 + m];            // B[ka+1][n]
    c = __builtin_amdgcn_wmma_f32_16x16x4_f32(false, a, false, b, (short)0, c,
                                              false, false);
  }
  int col = col0 + m;
  float bv = bias ? bias[col] : 0.f;
  #pragma unroll
  for (int i = 0; i < 8; ++i) {
    int r = row0 + i + hi * 8;             // C layout: VGPR i -> M=i (lanes 0-15) / M=8+i
    if (r < n) {
      float v = c[i] + bv;
      if (do_relu) v = fmaxf(v, 0.f);
      C[(size_t)r * 256 + col] = v;
    }
  }
}

// ---------------- readout + heads ----------------

__global__ __launch_bounds__(256) void colsum(const float* __restrict__ x,
                                              float* __restrict__ gsum, int n) {
  int j = threadIdx.x;
  int r0 = blockIdx.x * 64;
  int r1 = r0 + 64; if (r1 > n) r1 = n;
  float acc = 0.f;
  for (int r = r0; r < r1; ++r) acc += x[(size_t)r * 256 + j];
  atomicAdd(&gsum[j], acc);
}

// gpart[j] = actor_b1[j] + sum_k (gsum[k]/N) * actor_w1[256+k][j]
__global__ __launch_bounds__(256) void gpart_k(const float* __restrict__ gsum,
                                               const float* __restrict__ w1,
                                               const float* __restrict__ b1,
                                               float* __restrict__ gpart, int n) {
  int j = threadIdx.x;
  float acc = b1[j];
  float inv = 1.f / (float)n;
  for (int k = 0; k < 256; ++k) acc += gsum[k] * inv * w1[(size_t)(256 + k) * 256 + j];
  gpart[j] = acc;
}

// logits[n] = hidden[n][:] . w2 + b2
__global__ __launch_bounds__(256) void actor_out(const float* __restrict__ hidden,
                                                 const float* __restrict__ w2,
                                                 const float* __restrict__ b2,
                                                 float* __restrict__ out, int n) {
  int node = blockIdx.x * 8 + (threadIdx.x >> 5);
  if (node >= n) return;
  int lane = threadIdx.x & 31;
  const float* h = hidden + (size_t)node * 256;
  float acc = 0.f;
  for (int i = lane; i < 256; i += 32) acc += h[i] * w2[i];
  for (int off = 16; off > 0; off >>= 1) acc += __shfl_down(acc, off, 32);
  if (lane == 0) out[node] = acc + b2[0];
}

__global__ __launch_bounds__(256) void critic_k(
    const float* __restrict__ gsum,
    const float* __restrict__ w1, const float* __restrict__ b1,
    const float* __restrict__ w2, const float* __restrict__ b2,
    const float* __restrict__ w3, const float* __restrict__ b3,
    float* __restrict__ out, int n) {
  __shared__ float pooled[512];
  __shared__ float c1[512];
  __shared__ float c2[256];
  __shared__ float red[256];
  int t = threadIdx.x;
  float g = gsum[t] / (float)n;           // pooled = [g, g]
  pooled[t] = g; pooled[256 + t] = g;
  __syncthreads();
  for (int j = t; j < 512; j += 256) {
    float acc = b1[j];
    for (int k = 0; k < 512; ++k) acc += pooled[k] * w1[(size_t)k * 512 + j];
    c1[j] = fmaxf(acc, 0.f);
  }
  __syncthreads();
  {
    float acc = b2[t];
    for (int k = 0; k < 512; ++k) acc += c1[k] * w2[(size_t)k * 256 + t];
    c2[t] = fmaxf(acc, 0.f);
  }
  __syncthreads();
  red[t] = c2[t] * w3[t];
  __syncthreads();
  for (int s = 128; s > 0; s >>= 1) { if (t < s) red[t] += red[t + s]; __syncthreads(); }
  if (t == 0) out[0] = red[0] + b3[0];
}

// ---------------- launch ----------------

extern "C" void kernel_launch(void* const* d_in, const int* in_sizes, int n_in,
                              void* d_out, int out_size, void* d_ws, size_t ws_size,
                              hipStream_t stream) {
  const float* status  = (const float*)d_in[0];
  const float* est     = (const float*)d_in[1];
  const float* degf    = (const float*)d_in[2];
  const int*   esrc    = (const int*)d_in[3];
  const int*   edst    = (const int*)d_in[4];
  const int*   eid     = (const int*)d_in[5];
  const int*   epos    = (const int*)d_in[6];
  const float* etab    = (const float*)d_in[7];
  const float* att_w1  = (const float*)d_in[8];
  const float* att_b1  = (const float*)d_in[9];
  const float* att_w2  = (const float*)d_in[10];
  const float* att_b2  = (const float*)d_in[11];
  const float* conv1_w = (const float*)d_in[12];
  const float* conv1_b = (const float*)d_in[13];
  const float* conv2_w = (const float*)d_in[14];
  const float* conv2_b = (const float*)d_in[15];
  const float* aw1     = (const float*)d_in[16];
  const float* ab1     = (const float*)d_in[17];
  const float* aw2     = (const float*)d_in[18];
  const float* ab2     = (const float*)d_in[19];
  const float* cw1     = (const float*)d_in[20];
  const float* cb1     = (const float*)d_in[21];
  const float* cw2     = (const float*)d_in[22];
  const float* cb2     = (const float*)d_in[23];
  const float* cw3     = (const float*)d_in[24];
  const float* cb3     = (const float*)d_in[25];

  const int N = in_sizes[0];
  const int E = in_sizes[3];
  float* out = (float*)d_out;

  // workspace carve-up (floats)
  float* ws = (float*)d_ws;
  size_t off = 0;
  float* score = ws + off; off += (size_t)E;
  float* smax  = ws + off; off += (size_t)N;
  float* denom = ws + off; off += (size_t)N;
  float* agg   = ws + off; off += (size_t)N * 7;
  float* dinv  = ws + off; off += (size_t)N;      // deg, then rsqrt in place
  float* nf    = ws + off; off += (size_t)N * 10;
  float* red   = ws + off; off += 8;
  float* gsum  = ws + off; off += 256;
  float* gpart = ws + off; off += 256;
  off = (off + 63) & ~(size_t)63;                  // 256B-align the big buffers
  float* B1 = ws + off; off += (size_t)N * 256;
  float* B2 = ws + off; off += (size_t)N * 256;

  const int TB = 256;
  dim3 blkN((N + TB - 1) / TB);
  dim3 blkE((E + TB - 1) / TB);
  dim3 blkNE(((size_t)N * 256 + TB - 1) / TB);
  dim3 blkEdgeWave((E + 7) / 8);
  dim3 gemmGrid((N + 15) / 16, 2);

  // 1) init
  init_k<<<blkN, TB, 0, stream>>>(smax, denom, agg, dinv, red, gsum, N);
  // 2) edge attention
  edge_scores<<<blkE, TB, 0, stream>>>(eid, epos, etab, att_w1, att_b1, att_w2, att_b2,
                                       esrc, edst, score, smax, dinv, E);
  edge_expsum<<<blkE, TB, 0, stream>>>(score, esrc, edst, smax, denom, E);
  edge_agg<<<blkE, TB, 0, stream>>>(eid, epos, etab, score, esrc, edst, smax, denom, agg, E);
  // 3) GCN norm + node features
  dinv_k<<<blkN, TB, 0, stream>>>(dinv, N);
  reduce_stats<<<blkN, TB, 0, stream>>>(est, degf, red, N);
  node_feats_k<<<blkN, TB, 0, stream>>>(status, est, degf, agg, red, nf, N);
  // 4) GCN layer 1
  conv1_transform<<<dim3((N + 7) / 8), TB, 0, stream>>>(nf, conv1_w, B1, N);
  self_init<<<blkNE, TB, 0, stream>>>(B1, dinv, conv1_b, B2, N);
  gcn_aggregate<<<blkEdgeWave, TB, 0, stream>>>(B1, esrc, edst, dinv, B2, E);
  relu_k<<<blkNE, TB, 0, stream>>>(B2, (size_t)N * 256);
  // 5) GCN layer 2 (WMMA transform)
  gemm256_wmma<<<gemmGrid, TB, 0, stream>>>(B2, conv2_w, B1, nullptr, N, 0);
  self_init<<<blkNE, TB, 0, stream>>>(B1, dinv, conv2_b, B2, N);
  gcn_aggregate<<<blkEdgeWave, TB, 0, stream>>>(B1, esrc, edst, dinv, B2, E);
  relu_k<<<blkNE, TB, 0, stream>>>(B2, (size_t)N * 256);
  // 6) readout
  colsum<<<dim3((N + 63) / 64), TB, 0, stream>>>(B2, gsum, N);
  gpart_k<<<dim3(1), TB, 0, stream>>>(gsum, aw1, ab1, gpart, N);
  // 7) actor head: hidden = relu(x @ W1_top + gpart) via WMMA, then dot with w2
  gemm256_wmma<<<gemmGrid, TB, 0, stream>>>(B2, aw1, B1, gpart, N, 1);
  actor_out<<<dim3((N + 7) / 8), TB, 0, stream>>>(B1, aw2, ab2, out, N);
  // 8) critic head -> out[N]
  critic_k<<<dim3(1), TB, 0, stream>>>(gsum, cw1, cb1, cw2, cb2, cw3, cb3, out + N, N);

  (void)n_in; (void)out_size; (void)ws_size;
}